// CRSModel_75634374083155
// MI455X (gfx1250) — compile-verified
//
#include <hip/hip_runtime.h>

// ---------------------------------------------------------------------------
// CRS friction-model scan for MI455X (gfx1250, wave32).
//
// R' = R*et / (1 - (eta*R/sd)*(1-et))  is the Mobius map R' = A*R/(1+C*R),
// A = et, C = (et-1)*eta/sd, composing as (A1,C1)∘(A2,C2) = (A2A1, C2A1+C1).
//
// One wave32 per row (8192 waves). 128 timesteps per chunk:
//   * lane l owns t = c*128 + 4l .. 4l+3  -> aligned global_load_b128 (NT)
//   * lane locally composes its 4 maps (3 ops each), then a single 5-step
//     shfl_up scan composes all 128 -> DS cost amortized 4x vs 32/chunk
//   * exclusive prefix applied to carried R gives lane's exact R_{t-1};
//     4 sequential steps reproduce the reference denom / R_t / N_t
//   * 5-step prefix sum of per-lane N totals builds the Nt cumsum
//   * carries broadcast from lane 31 via v_readlane (no DS op)
//
// Roofline: ~20 B/elem * 33.5M elems ~= 670 MB -> ~29 us @ 23.3 TB/s.
// After 4x scan amortization, VALU/elem ~1 + 2 transcendentals -> memory
// bound. NT loads/stores keep the 670 MB/pass stream from thrashing the
// 192 MB L2; global_prefetch_b8 runs ~2 chunks (1 KB/stream) ahead.
// ---------------------------------------------------------------------------

#define C_TNSR  0.001f
#define C_TSSR  0.002f
#define C_SIGMA 50.0f
#define C_BIOT  0.3f
#define C_R0    0.0001f
#define C_N0    (0.0001f * 1.0f)   // R0 * INIT_DT

// Native clang vector (16B aligned) so __builtin_nontemporal_load accepts it.
typedef float v4f __attribute__((ext_vector_type(4)));

// v_rcp_f32 (~1 ulp) + one NR step (~0.5 ulp): 3 VALU ops vs ~12-op IEEE div.
__device__ __forceinline__ float fast_rcp(float x) {
    float r = __builtin_amdgcn_rcpf(x);
    r = __fmaf_rn(__fmaf_rn(-x, r, 1.0f), r, r);
    return r;
}

__device__ __forceinline__ float bcast31(float x) {
    return __int_as_float(__builtin_amdgcn_readlane(__float_as_int(x), 31));
}

__global__ __launch_bounds__(256) void crs_wave_scan_kernel(
    const float* __restrict__ params,   // (B,3): mu, rc, rf
    const float* __restrict__ p,        // (B,T)
    const float* __restrict__ dpdt,     // (B,T)
    const float* __restrict__ dt,       // (B,T)
    float* __restrict__ Rt,             // (B,T+1)
    float* __restrict__ Nt,             // (B,T+1)
    int B, int T)
{
    const int wavesPerBlock = blockDim.x >> 5;
    const int b    = blockIdx.x * wavesPerBlock + (threadIdx.x >> 5); // row = wave
    const int lane = threadIdx.x & 31;
    if (b >= B) return;                       // wave-uniform exit

    const float mu  = params[b * 3 + 0];
    const float rc  = params[b * 3 + 1];
    const float rf  = params[b * 3 + 2];
    const float eta = 1.0f / rf;              // once per row
    const float s0  = C_TSSR - mu * C_TNSR;   // s_dot = fma(mu, dpdt, s0)
    const float a0  = rc * C_SIGMA;           // asig  = fma(a1, p, a0)
    const float a1  = -(rc * C_BIOT);

    const size_t rowIn  = (size_t)b * (size_t)T;
    const size_t rowOut = (size_t)b * (size_t)(T + 1);

    if (lane == 0) {                          // Rt[:,0] = R0 ; Nt[:,0] = N0
        Rt[rowOut] = C_R0;
        Nt[rowOut] = C_N0;
    }

    float Rc   = C_R0;                        // carried recurrence state
    float Ncum = C_N0;                        // carried cumulative N

    const int chunks = T >> 7;                // 128 timesteps / chunk
    for (int c = 0; c < chunks; ++c) {
        const size_t idx4 = rowIn + ((size_t)c << 7) + ((size_t)lane << 2);

        // Stream ~2 chunks ahead per input (global_prefetch_b8 on gfx1250).
        if (c + 2 < chunks) {
            __builtin_prefetch(&p[idx4 + 256],    0, 1);
            __builtin_prefetch(&dpdt[idx4 + 256], 0, 1);
            __builtin_prefetch(&dt[idx4 + 256],   0, 1);
        }

        // Aligned, coalesced, non-temporal b128 loads (read-once stream).
        const v4f p4 = __builtin_nontemporal_load(
            reinterpret_cast<const v4f*>(p + idx4));
        const v4f d4 = __builtin_nontemporal_load(
            reinterpret_cast<const v4f*>(dpdt + idx4));
        const v4f t4 = __builtin_nontemporal_load(
            reinterpret_cast<const v4f*>(dt + idx4));

        // Elementwise terms for the lane's 4 timesteps.
        float et[4], Cm[4], asigrf[4];
        {
            const float pv[4] = {p4.x, p4.y, p4.z, p4.w};
            const float dv[4] = {d4.x, d4.y, d4.z, d4.w};
            const float tv[4] = {t4.x, t4.y, t4.z, t4.w};
            #pragma unroll
            for (int i = 0; i < 4; ++i) {
                const float sd   = __fmaf_rn(mu, dv[i], s0);          // s_dot
                const float asig = __fmaf_rn(a1, pv[i], a0);          // rc*(SIGMA-BIOT*p)
                et[i]     = __expf(sd * tv[i] * fast_rcp(asig));      // exp_term
                Cm[i]     = (et[i] - 1.0f) * eta * fast_rcp(sd);      // Mobius C
                asigrf[i] = asig * rf;                                // asig/eta
            }
        }

        // Compose the lane's 4 maps (time order), then wave inclusive scan.
        float A = et[0], Cc = Cm[0];
        #pragma unroll
        for (int i = 1; i < 4; ++i) {
            Cc = __fmaf_rn(Cm[i], A, Cc);
            A  = et[i] * A;
        }
        #pragma unroll
        for (int d = 1; d < 32; d <<= 1) {
            const float Ap = __shfl_up(A, d, 32);
            const float Cp = __shfl_up(Cc, d, 32);
            if (lane >= d) {
                Cc = __fmaf_rn(Cc, Ap, Cp);
                A  = A * Ap;
            }
        }

        // Exclusive prefix -> R before the lane's first step.
        float Aex = __shfl_up(A, 1, 32);
        float Cex = __shfl_up(Cc, 1, 32);
        if (lane == 0) { Aex = 1.0f; Cex = 0.0f; }
        float R = (Aex * Rc) * fast_rcp(__fmaf_rn(Cex, Rc, 1.0f));

        // Reconstruct the 4 steps exactly as the reference:
        //   denom = 1 + C*R ;  R' = R*et/denom ;  N = (asig/eta)*log(denom)
        const size_t o = rowOut + ((size_t)c << 7) + ((size_t)lane << 2) + 1;
        float N[4], localSum = 0.0f;
        #pragma unroll
        for (int i = 0; i < 4; ++i) {
            const float denom = __fmaf_rn(Cm[i], R, 1.0f);
            R    = R * et[i] * fast_rcp(denom);
            N[i] = asigrf[i] * __logf(denom);
            localSum += N[i];
            __builtin_nontemporal_store(R, &Rt[o + i]);
        }

        // Wave prefix sum of per-lane N totals -> cumsum base per lane.
        float Ns = localSum;
        #pragma unroll
        for (int d = 1; d < 32; d <<= 1) {
            const float t = __shfl_up(Ns, d, 32);
            if (lane >= d) Ns += t;
        }
        float run = Ncum + (Ns - localSum);   // exclusive prefix for this lane
        #pragma unroll
        for (int i = 0; i < 4; ++i) {
            run += N[i];
            __builtin_nontemporal_store(run, &Nt[o + i]);
        }

        // Carries from lane 31 via v_readlane (all lanes active).
        Rc    = bcast31(R);
        Ncum += bcast31(Ns);
    }
}

// Generic fallback (any T): one thread per row, plain sequential scan.
__global__ void crs_row_seq_kernel(
    const float* __restrict__ params, const float* __restrict__ p,
    const float* __restrict__ dpdt, const float* __restrict__ dt,
    float* __restrict__ Rt, float* __restrict__ Nt, int B, int T)
{
    const int b = blockIdx.x * blockDim.x + threadIdx.x;
    if (b >= B) return;
    const float mu = params[b*3+0], rc = params[b*3+1], rf = params[b*3+2];
    const float eta = 1.0f / rf;
    const size_t rowIn = (size_t)b * T, rowOut = (size_t)b * (T + 1);
    float R = C_R0, Ncum = C_N0;
    Rt[rowOut] = C_R0; Nt[rowOut] = C_N0;
    for (int t = 0; t < T; ++t) {
        const float sd   = C_TSSR - mu * (C_TNSR - dpdt[rowIn + t]);
        const float asig = rc * (C_SIGMA - C_BIOT * p[rowIn + t]);
        const float et   = __expf(sd * dt[rowIn + t] / asig);
        const float den  = 1.0f - (eta * R / sd) * (1.0f - et);
        R = R * et / den;
        Ncum += (asig / eta) * __logf(den);
        Rt[rowOut + t + 1] = R;
        Nt[rowOut + t + 1] = Ncum;
    }
}

extern "C" void kernel_launch(void* const* d_in, const int* in_sizes, int n_in,
                              void* d_out, int out_size, void* d_ws, size_t ws_size,
                              hipStream_t stream) {
    (void)n_in; (void)d_ws; (void)ws_size; (void)out_size;

    const float* params = (const float*)d_in[0];   // (B,3)
    const float* p      = (const float*)d_in[1];   // (B,T)
    const float* dpdt   = (const float*)d_in[2];   // (B,T)
    const float* dtv    = (const float*)d_in[3];   // (B,T)

    const int B = in_sizes[0] / 3;
    const int T = in_sizes[1] / B;                 // 4096

    float* Rt = (float*)d_out;                     // (B, T+1)
    float* Nt = Rt + (size_t)B * (size_t)(T + 1);  // (B, T+1)

    if ((T & 127) == 0) {
        const int WAVES_PER_BLOCK = 8;             // 256 threads = 8 wave32
        const int blocks = (B + WAVES_PER_BLOCK - 1) / WAVES_PER_BLOCK;
        crs_wave_scan_kernel<<<blocks, WAVES_PER_BLOCK * 32, 0, stream>>>(
            params, p, dpdt, dtv, Rt, Nt, B, T);
    } else {
        const int threads = 256;
        const int blocks = (B + threads - 1) / threads;
        crs_row_seq_kernel<<<blocks, threads, 0, stream>>>(
            params, p, dpdt, dtv, Rt, Nt, B, T);
    }
}